// MyOutGCN_687194767719
// MI455X (gfx1250) — compile-verified
//
#include <hip/hip_runtime.h>
#include <math.h>

#define N_NODES 100000
#define N_EDGES 1250000
#define K_SEL   20000
#define DIM_IN  128
#define DIM_H   64
#define DIM_OUT 5
#define NEG_SLOPE 0.01f
#define BN_EPS    1e-5f

typedef float v2f __attribute__((ext_vector_type(2)));
typedef float v8f __attribute__((ext_vector_type(8)));

// ---------------- utility ----------------
__global__ void zero_f32(float* __restrict__ p, int n) {
    int i = blockIdx.x * blockDim.x + threadIdx.x;
    if (i < n) p[i] = 0.0f;
}

// deg[col[e]] += 1 for each directed edge
__global__ void deg_accum(const int* __restrict__ col, float* __restrict__ deg, int e_cnt) {
    int i = blockIdx.x * blockDim.x + threadIdx.x;
    if (i < e_cnt) atomicAdd(&deg[col[i]], 1.0f);
}

// add self-loop, convert to D^{-1/2}
__global__ void dis_finalize(float* __restrict__ deg_dis, int n) {
    int i = blockIdx.x * blockDim.x + threadIdx.x;
    if (i < n) {
        float d = deg_dis[i] + 1.0f;   // self loop => deg >= 1 always
        deg_dis[i] = rsqrtf(d);
    }
}

// per-edge symmetric norm, computed once, reused by both conv layers
__global__ void edge_norm(const int* __restrict__ row, const int* __restrict__ col,
                          const float* __restrict__ dis, float* __restrict__ w, int e_cnt) {
    int e = blockIdx.x * blockDim.x + threadIdx.x;
    if (e < e_cnt) w[e] = dis[row[e]] * dis[col[e]];
}

// ---------------- f32 WMMA GEMM: C[M,NC] = A[M,KDIM] @ B[KDIM,NC] ----------------
// One wave per 16-row strip covering all NC columns (NT accumulators).
// B is staged once per block into LDS, pre-swizzled into WMMA fragment layout so the
// inner loop does one conflict-free ds_load_b64 per column tile.
template <int KDIM, int NT>   // NT = NC/16 column tiles
__global__ void gemm_wmma_f32(const float* __restrict__ A, const float* __restrict__ B,
                              float* __restrict__ C, int M) {
    constexpr int NC    = NT * 16;
    constexpr int PAIRS = (KDIM / 4) * NT * 32;      // one v2f fragment per (kstep,tile,lane)
    __shared__ float sB[KDIM * NC];                  // = PAIRS * 2 floats

    const int tid = threadIdx.x;

    // ---- stage B into fragment-swizzled LDS ----
    // pair j <-> (kstep ks, tile t, lane ln): lane ln needs B[(4ks+2h)+{0,1}][16t+l]
    for (int j = tid; j < PAIRS; j += blockDim.x) {
        int ks  = j / (NT * 32);
        int rem = j % (NT * 32);
        int t   = rem / 32;
        int ln  = rem % 32;
        int h = ln >> 4, l = ln & 15;
        int src = (4 * ks + 2 * h) * NC + 16 * t + l;
        sB[2 * j]     = B[src];
        sB[2 * j + 1] = B[src + NC];
    }
    __syncthreads();                                  // all waves participate (no early return)

    const int wave = tid >> 5;
    const int lane = tid & 31;
    const int m0   = (blockIdx.x * (blockDim.x >> 5) + wave) << 4;

    if (m0 < M) {                                     // uniform per wave; EXEC all-1s for WMMA
        const int half = lane >> 4;                   // 0: lanes 0-15, 1: lanes 16-31
        const int l    = lane & 15;
        const float* Ar = A + (size_t)(m0 + l) * KDIM;   // M % 16 == 0 here

        v8f acc[NT];
        #pragma unroll
        for (int t = 0; t < NT; ++t) acc[t] = (v8f){};

        #pragma unroll 2
        for (int ks = 0; ks < KDIM / 4; ++ks) {
            // A 16x4 layout: VGPR0 holds K = 4ks+2h, VGPR1 holds K = 4ks+2h+1
            v2f a;
            a.x = Ar[4 * ks + 2 * half];
            a.y = Ar[4 * ks + 2 * half + 1];
            #pragma unroll
            for (int t = 0; t < NT; ++t) {
                v2f b = *reinterpret_cast<const v2f*>(&sB[((ks * NT + t) * 32 + lane) * 2]);
                acc[t] = __builtin_amdgcn_wmma_f32_16x16x4_f32(
                            /*neg_a=*/false, a, /*neg_b=*/false, b,
                            /*c_mod=*/(short)0, acc[t],
                            /*reuse_a=*/false, /*reuse_b=*/false);
            }
        }
        // C/D layout: VGPR r -> M = m0 + r + 8*half, N = 16*t + l
        #pragma unroll
        for (int t = 0; t < NT; ++t) {
            #pragma unroll
            for (int r = 0; r < 8; ++r) {
                int rowM = m0 + r + 8 * half;
                C[(size_t)rowM * NC + 16 * t + l] = acc[t][r];
            }
        }
    }
}

// ---------------- conv accumulate buffer: bias + self-loop term ----------------
__global__ void conv_init(const float* __restrict__ hw, const float* __restrict__ dis,
                          const float* __restrict__ bias, float* __restrict__ acc, int total) {
    int i = blockIdx.x * blockDim.x + threadIdx.x;
    if (i < total) {
        int node = i >> 6, f = i & 63;
        float d = dis[node];
        acc[i] = bias[f] + hw[i] * d * d;     // self-loop norm = dis[i]^2
    }
}

// ---------------- edge scatter: acc[col] += hw[row] * w[e] ----------------
// 16 threads per edge, float4 per thread (64 features)
__global__ void edge_scatter(const int* __restrict__ row, const int* __restrict__ col,
                             const float* __restrict__ w, const float* __restrict__ hw,
                             float* __restrict__ acc, int e_cnt) {
    int gid = blockIdx.x * blockDim.x + threadIdx.x;
    int e = gid >> 4;
    if (e >= e_cnt) return;
    int f = (gid & 15) << 2;
    int r = row[e], c = col[e];
    float we = w[e];
    const float4 v = *reinterpret_cast<const float4*>(hw + (size_t)r * 64 + f);
    float* dst = acc + (size_t)c * 64 + f;
    atomicAdd(dst + 0, v.x * we);
    atomicAdd(dst + 1, v.y * we);
    atomicAdd(dst + 2, v.z * we);
    atomicAdd(dst + 3, v.w * we);
}

// ---------------- leaky-relu + per-channel sum / sumsq (64 channels) ----------------
// requires gridDim.x*blockDim.x % 64 == 0 so each thread's channel is fixed
__global__ void act_stats(const float* __restrict__ in, float* __restrict__ out,
                          float* __restrict__ stats, int total) {
    __shared__ float s_sum[64];
    __shared__ float s_sq[64];
    int tid = threadIdx.x;
    if (tid < 64) { s_sum[tid] = 0.0f; s_sq[tid] = 0.0f; }
    __syncthreads();

    int gid = blockIdx.x * blockDim.x + tid;
    int stride = gridDim.x * blockDim.x;
    int f = gid & 63;
    float sum = 0.0f, sq = 0.0f;
    for (int i = gid; i < total; i += stride) {
        float v = in[i];
        v = (v > 0.0f) ? v : NEG_SLOPE * v;
        out[i] = v;
        sum += v;
        sq  += v * v;
    }
    atomicAdd(&s_sum[f], sum);
    atomicAdd(&s_sq[f],  sq);
    __syncthreads();
    if (tid < 64) {
        atomicAdd(&stats[tid],      s_sum[tid]);
        atomicAdd(&stats[64 + tid], s_sq[tid]);
    }
}

// stats[0..63]=sum, [64..127]=sumsq  ->  [128..191]=scale, [192..255]=shift
__global__ void bn_finalize(const float* __restrict__ gamma, const float* __restrict__ beta,
                            float* __restrict__ stats, int n_nodes) {
    int f = threadIdx.x;
    if (f < 64) {
        float inv_n = 1.0f / (float)n_nodes;
        float mu  = stats[f] * inv_n;
        float var = stats[64 + f] * inv_n - mu * mu;
        var = var > 0.0f ? var : 0.0f;
        float scale = gamma[f] * rsqrtf(var + BN_EPS);
        stats[128 + f] = scale;
        stats[192 + f] = beta[f] - mu * scale;
    }
}

__global__ void bn_apply(const float* __restrict__ in, const float* __restrict__ stats,
                         float* __restrict__ out, int total) {
    int i = blockIdx.x * blockDim.x + threadIdx.x;
    if (i < total) {
        int f = i & 63;
        out[i] = in[i] * stats[128 + f] + stats[192 + f];
    }
}

// ---------------- select rows into output ----------------
__global__ void gather_sel(const float* __restrict__ h, const int* __restrict__ idx,
                           float* __restrict__ out, int total /* K*64 */) {
    int i = blockIdx.x * blockDim.x + threadIdx.x;
    if (i < total) {
        int k = i >> 6, f = i & 63;
        out[i] = h[(size_t)idx[k] * 64 + f];
    }
}

// ---------------- final tiny MLP + sigmoid ----------------
__global__ void mlp_sigmoid(const float* __restrict__ h, const int* __restrict__ idx,
                            const float* __restrict__ Wm, const float* __restrict__ bm,
                            float* __restrict__ out, int kcnt) {
    int gid = blockIdx.x * blockDim.x + threadIdx.x;
    int k = gid / DIM_OUT, j = gid % DIM_OUT;
    if (k >= kcnt) return;
    const float* hr = h + (size_t)idx[k] * 64;
    float z = bm[j];
    #pragma unroll 8
    for (int f = 0; f < 64; ++f) z += hr[f] * Wm[f * DIM_OUT + j];
    out[(size_t)k * DIM_OUT + j] = 1.0f / (1.0f + expf(-z));
}

// ===================== host launcher =====================
extern "C" void kernel_launch(void* const* d_in, const int* in_sizes, int n_in,
                              void* d_out, int out_size, void* d_ws, size_t ws_size,
                              hipStream_t stream) {
    const float* x      = (const float*)d_in[0];
    const int*   ei     = (const int*)  d_in[1];
    const int*   idx    = (const int*)  d_in[2];
    const float* W0     = (const float*)d_in[3];
    const float* b0     = (const float*)d_in[4];
    const float* gamma0 = (const float*)d_in[5];
    const float* beta0  = (const float*)d_in[6];
    const float* W1     = (const float*)d_in[7];
    const float* b1     = (const float*)d_in[8];
    const float* gamma1 = (const float*)d_in[9];
    const float* beta1  = (const float*)d_in[10];
    const float* Wm     = (const float*)d_in[11];
    const float* bm     = (const float*)d_in[12];

    const int* row = ei;             // edge_index[0] = source
    const int* col = ei + N_EDGES;   // edge_index[1] = target

    const int NH = N_NODES * DIM_H;  // 6,400,000
    float* ws    = (float*)d_ws;
    float* dis   = ws;               // [N]
    float* buf1  = dis  + N_NODES;   // [N*64]
    float* buf2  = buf1 + NH;        // [N*64]
    float* buf3  = buf2 + NH;        // [N*64]
    float* stats = buf3 + NH;        // [512] : 2 BN layers x 256
    float* wnorm = stats + 512;      // [E]

    float* out_hsel = (float*)d_out;                     // [K*64]
    float* out_mlp  = out_hsel + (size_t)K_SEL * DIM_H;  // [K*5]

    const int B = 256;
    auto blk = [](int n, int b) { return (n + b - 1) / b; };

    // degrees -> D^{-1/2} -> per-edge norm
    zero_f32<<<blk(N_NODES, B), B, 0, stream>>>(dis, N_NODES);
    zero_f32<<<blk(512, B), B, 0, stream>>>(stats, 512);
    deg_accum<<<blk(N_EDGES, B), B, 0, stream>>>(col, dis, N_EDGES);
    dis_finalize<<<blk(N_NODES, B), B, 0, stream>>>(dis, N_NODES);
    edge_norm<<<blk(N_EDGES, B), B, 0, stream>>>(row, col, dis, wnorm, N_EDGES);

    const int strips      = N_NODES / 16;        // 6250 row strips, 8 per block
    const int gemm_blocks = blk(strips, B / 32); // 782

    // ---- layer 0 ----
    gemm_wmma_f32<DIM_IN, DIM_H / 16><<<gemm_blocks, B, 0, stream>>>(x, W0, buf1, N_NODES);
    conv_init<<<blk(NH, B), B, 0, stream>>>(buf1, dis, b0, buf2, NH);
    edge_scatter<<<blk(N_EDGES * 16, B), B, 0, stream>>>(row, col, wnorm, buf1, buf2, N_EDGES);
    act_stats<<<1024, B, 0, stream>>>(buf2, buf3, stats, NH);
    bn_finalize<<<1, 64, 0, stream>>>(gamma0, beta0, stats, N_NODES);
    bn_apply<<<blk(NH, B), B, 0, stream>>>(buf3, stats, buf1, NH);

    // ---- layer 1 ----
    gemm_wmma_f32<DIM_H, DIM_H / 16><<<gemm_blocks, B, 0, stream>>>(buf1, W1, buf2, N_NODES);
    conv_init<<<blk(NH, B), B, 0, stream>>>(buf2, dis, b1, buf3, NH);
    edge_scatter<<<blk(N_EDGES * 16, B), B, 0, stream>>>(row, col, wnorm, buf2, buf3, N_EDGES);
    act_stats<<<1024, B, 0, stream>>>(buf3, buf1, stats + 256, NH);
    bn_finalize<<<1, 64, 0, stream>>>(gamma1, beta1, stats + 256, N_NODES);
    bn_apply<<<blk(NH, B), B, 0, stream>>>(buf1, stats + 256, buf2, NH);

    // ---- outputs ----
    gather_sel<<<blk(K_SEL * DIM_H, B), B, 0, stream>>>(buf2, idx, out_hsel, K_SEL * DIM_H);
    mlp_sigmoid<<<blk(K_SEL * DIM_OUT, B), B, 0, stream>>>(buf2, idx, Wm, bm, out_mlp, K_SEL);
}